// Multi_head_focus_attention_48919677501896
// MI455X (gfx1250) — compile-verified
//
#include <hip/hip_runtime.h>
#include <hip/hip_bf16.h>

typedef __attribute__((ext_vector_type(2))) float v2f;
typedef __attribute__((ext_vector_type(8))) float v8f;

#define BATCH 4
#define CIN   256
#define HDIM  128
#define WDIM  128
#define HW    16384       // 128*128
#define NHEAD 8
#define DK    64
#define OQ    512         // NHEAD*DK
#define K7    7
#define RAD   3
#define TK    22          // 16 + 2*RAD halo
#define LSTR  20          // LDS row stride (floats): mult of 4 -> 16B-aligned b128 reads

// ---------------------------------------------------------------------------
// Kernel 1: fused QKV projection.  Y[o,p] = sum_c W[o,c] * X[c,p]
// One wave computes a 16(o) x 64(p) tile via V_WMMA_F32_16X16X4_F32,
// 4-way register blocking along pixels (A tile reused 4x per K-step).
// grid = (HW/64, OQ/64, BATCH*3), block = 128 (4 waves; wave -> o-subtile)
// ---------------------------------------------------------------------------
__global__ __launch_bounds__(128) void proj_kernel(
    const float* __restrict__ q, const float* __restrict__ k, const float* __restrict__ v,
    const float* __restrict__ Wq, const float* __restrict__ Wk, const float* __restrict__ Wv,
    float* __restrict__ qh, float* __restrict__ kh, float* __restrict__ vh)
{
  const int wave = threadIdx.x >> 5;
  const int lane = threadIdx.x & 31;
  const int lm   = lane & 15;       // M (A) / N (B,C,D) index
  const int hi   = lane >> 4;       // half-wave select
  const int kh2  = hi << 1;         // K base within 4-wide K step (0 or 2)

  const int b   = blockIdx.z / 3;
  const int mat = blockIdx.z % 3;
  const float* X  = (mat == 0) ? q  : (mat == 1) ? k  : v;
  const float* Wm = (mat == 0) ? Wq : (mat == 1) ? Wk : Wv;
  float*       Y  = (mat == 0) ? qh : (mat == 1) ? kh : vh;
  X += (size_t)b * CIN * HW;
  Y += (size_t)b * OQ  * HW;

  const int o0 = (blockIdx.y * 4 + wave) * 16;
  const int p0 = blockIdx.x * 64;

  v8f acc0 = {}; v8f acc1 = {}; v8f acc2 = {}; v8f acc3 = {};
  for (int c0 = 0; c0 < CIN; c0 += 4) {
    // A: W[o0+lm, c0+kh2 .. +1]  (contiguous -> b64 load)
    v2f a = *(const v2f*)(Wm + (size_t)(o0 + lm) * CIN + c0 + kh2);
    // B: X[c0+kh2, p] / X[c0+kh2+1, p]
    const float* Xb = X + (size_t)(c0 + kh2) * HW + p0 + lm;
    v2f b0, b1, b2, b3;
    b0.x = Xb[0];  b0.y = Xb[HW];
    b1.x = Xb[16]; b1.y = Xb[HW + 16];
    b2.x = Xb[32]; b2.y = Xb[HW + 32];
    b3.x = Xb[48]; b3.y = Xb[HW + 48];
    acc0 = __builtin_amdgcn_wmma_f32_16x16x4_f32(false, a, false, b0, (short)0, acc0, false, false);
    acc1 = __builtin_amdgcn_wmma_f32_16x16x4_f32(false, a, false, b1, (short)0, acc1, false, false);
    acc2 = __builtin_amdgcn_wmma_f32_16x16x4_f32(false, a, false, b2, (short)0, acc2, false, false);
    acc3 = __builtin_amdgcn_wmma_f32_16x16x4_f32(false, a, false, b3, (short)0, acc3, false, false);
  }
#pragma unroll
  for (int r = 0; r < 8; ++r) {
    const int m = r + (hi << 3);
    float* Yr = Y + (size_t)(o0 + m) * HW + p0 + lm;
    Yr[0]  = acc0[r];
    Yr[16] = acc1[r];
    Yr[32] = acc2[r];
    Yr[48] = acc3[r];
  }
}

// ---------------------------------------------------------------------------
// Kernel 2: local 7x7 attention per (batch, head, 16x16 pixel tile).
// One thread owns one pixel. k/v tiles (22x22 halo x 16 d-chunk) staged in LDS.
// corr[49] kept in registers (fully unrolled), in-thread softmax, then
// attention-weighted v accumulation chunk by chunk.
// grid = (W/16, H/16, BATCH*NHEAD), block = 256
// ---------------------------------------------------------------------------
__global__ __launch_bounds__(256) void attn_kernel(
    const float* __restrict__ qh, const float* __restrict__ kh,
    const float* __restrict__ vh, float* __restrict__ ao)
{
  __shared__ float sk[TK * TK * LSTR];   // 38,720 B

  const int tid = threadIdx.x;
  const int tx  = tid & 15;
  const int ty  = tid >> 4;
  const int w0  = blockIdx.x * 16;
  const int h0  = blockIdx.y * 16;
  const int b   = blockIdx.z >> 3;
  const int n   = blockIdx.z & 7;
  const size_t base = ((size_t)b * OQ + (size_t)n * DK) * HW;
  const int p = (h0 + ty) * WDIM + (w0 + tx);

  float corr[49];
#pragma unroll
  for (int i = 0; i < 49; ++i) corr[i] = 0.f;

  // ---- correlation: accumulate over 4 chunks of 16 d ----
  for (int dc = 0; dc < 4; ++dc) {
    const int d0 = dc * 16;
    float qreg[16];
#pragma unroll
    for (int dd = 0; dd < 16; ++dd)
      qreg[dd] = qh[base + (size_t)(d0 + dd) * HW + p];

    __syncthreads();
    for (int idx = tid; idx < TK * TK * 16; idx += 256) {
      const int dd  = idx / (TK * TK);
      const int pix = idx - dd * (TK * TK);
      const int hh  = h0 - RAD + pix / TK;
      const int ww  = w0 - RAD + pix % TK;
      float val = 0.f;
      if (hh >= 0 && hh < HDIM && ww >= 0 && ww < WDIM)
        val = kh[base + (size_t)(d0 + dd) * HW + hh * WDIM + ww];
      sk[pix * LSTR + dd] = val;
    }
    __syncthreads();

#pragma unroll
    for (int dy = 0; dy < K7; ++dy)
#pragma unroll
      for (int dx = 0; dx < K7; ++dx) {
        const float* kp = &sk[((ty + dy) * TK + (tx + dx)) * LSTR];
        float s = corr[dy * K7 + dx];
#pragma unroll
        for (int dd = 0; dd < 16; ++dd) s += qreg[dd] * kp[dd];
        corr[dy * K7 + dx] = s;
      }
  }

  // ---- softmax over 49 offsets, temperature sqrt(64)=8 ----
  float mx = corr[0];
#pragma unroll
  for (int i = 1; i < 49; ++i) mx = fmaxf(mx, corr[i]);
  float sum = 0.f;
#pragma unroll
  for (int i = 0; i < 49; ++i) {
    corr[i] = __expf((corr[i] - mx) * 0.125f);
    sum += corr[i];
  }
  const float inv = 1.0f / sum;
#pragma unroll
  for (int i = 0; i < 49; ++i) corr[i] *= inv;

  // ---- weighted sum of v, chunk by chunk ----
  for (int dc = 0; dc < 4; ++dc) {
    const int d0 = dc * 16;
    __syncthreads();
    for (int idx = tid; idx < TK * TK * 16; idx += 256) {
      const int dd  = idx / (TK * TK);
      const int pix = idx - dd * (TK * TK);
      const int hh  = h0 - RAD + pix / TK;
      const int ww  = w0 - RAD + pix % TK;
      float val = 0.f;
      if (hh >= 0 && hh < HDIM && ww >= 0 && ww < WDIM)
        val = vh[base + (size_t)(d0 + dd) * HW + hh * WDIM + ww];
      sk[pix * LSTR + dd] = val;
    }
    __syncthreads();

    float o[16];
#pragma unroll
    for (int dd = 0; dd < 16; ++dd) o[dd] = 0.f;
#pragma unroll
    for (int dy = 0; dy < K7; ++dy)
#pragma unroll
      for (int dx = 0; dx < K7; ++dx) {
        const float wgt = corr[dy * K7 + dx];
        const float* vp = &sk[((ty + dy) * TK + (tx + dx)) * LSTR];
#pragma unroll
        for (int dd = 0; dd < 16; ++dd) o[dd] += wgt * vp[dd];
      }
#pragma unroll
    for (int dd = 0; dd < 16; ++dd)
      ao[base + (size_t)(d0 + dd) * HW + p] = o[dd];
  }
}

// ---------------------------------------------------------------------------
// Kernel 3: FC projection (WMMA f32) + residual + LayerNorm over channels.
// One block = 16 pixels x all 256 output channels (8 waves x two 16x16 tiles,
// K = 512).  Results staged in LDS; per-pixel mean/var by 16-lane shfl tree.
// grid = (HW/16, BATCH), block = 256
// ---------------------------------------------------------------------------
__global__ __launch_bounds__(256) void fc_ln_kernel(
    const float* __restrict__ ao, const float* __restrict__ Wfc,
    const float* __restrict__ qin, const float* __restrict__ lnw,
    const float* __restrict__ lnb, float* __restrict__ out)
{
  __shared__ float sfc[CIN * 17];   // [co][px], padded
  __shared__ float smu[16];
  __shared__ float srs[16];

  const int tid  = threadIdx.x;
  const int wave = tid >> 5;
  const int lane = tid & 31;
  const int lm   = lane & 15;
  const int hi   = lane >> 4;
  const int kh2  = hi << 1;

  const int b  = blockIdx.y;
  const int p0 = blockIdx.x * 16;
  const float* AO = ao  + (size_t)b * OQ  * HW;
  const float* Q  = qin + (size_t)b * CIN * HW;
  float*       O  = out + (size_t)b * CIN * HW;

  const int co0 = wave * 32;
  v8f acc0 = {}; v8f acc1 = {};
  for (int ci = 0; ci < OQ; ci += 4) {
    const float* Xb = AO + (size_t)(ci + kh2) * HW + p0 + lm;
    v2f bb; bb.x = Xb[0]; bb.y = Xb[HW];
    v2f a0 = *(const v2f*)(Wfc + (size_t)(co0      + lm) * OQ + ci + kh2);
    v2f a1 = *(const v2f*)(Wfc + (size_t)(co0 + 16 + lm) * OQ + ci + kh2);
    acc0 = __builtin_amdgcn_wmma_f32_16x16x4_f32(false, a0, false, bb, (short)0, acc0, false, false);
    acc1 = __builtin_amdgcn_wmma_f32_16x16x4_f32(false, a1, false, bb, (short)0, acc1, false, false);
  }
  // residual add + stage to LDS
#pragma unroll
  for (int r = 0; r < 8; ++r) {
    const int m  = r + (hi << 3);
    const int co = co0 + m;
    sfc[co * 17 + lm]        = acc0[r] + Q[(size_t)co * HW + p0 + lm];
    sfc[(co + 16) * 17 + lm] = acc1[r] + Q[(size_t)(co + 16) * HW + p0 + lm];
  }
  __syncthreads();

  // LayerNorm stats: 16 lanes per pixel, each sums 16 channels, shfl reduce
  const int px = tid >> 4;
  const int j  = tid & 15;
  float sum = 0.f, sumsq = 0.f;
#pragma unroll
  for (int g = 0; g < 16; ++g) {
    const float x = sfc[(g * 16 + j) * 17 + px];
    sum += x; sumsq += x * x;
  }
#pragma unroll
  for (int m = 8; m >= 1; m >>= 1) {
    sum   += __shfl_xor(sum,   m, 32);
    sumsq += __shfl_xor(sumsq, m, 32);
  }
  const float mu  = sum * (1.0f / 256.0f);
  const float var = sumsq * (1.0f / 256.0f) - mu * mu;
  const float rs  = rsqrtf(var + 1e-6f);
  if (j == 0) { smu[px] = mu; srs[px] = rs; }
  __syncthreads();

  // coalesced normalize + affine + store
  for (int idx = tid; idx < CIN * 16; idx += 256) {
    const int co  = idx >> 4;
    const int pxx = idx & 15;
    const float x = sfc[co * 17 + pxx];
    const float y = lnw[co] * ((x - smu[pxx]) * srs[pxx]) + lnb[co];
    O[(size_t)co * HW + p0 + pxx] = y;
  }
}

// ---------------------------------------------------------------------------
// Host-side launcher.  Workspace: qh | kh | vh | attn_out  (4 x 128 MiB f32)
// ---------------------------------------------------------------------------
extern "C" void kernel_launch(void* const* d_in, const int* in_sizes, int n_in,
                              void* d_out, int out_size, void* d_ws, size_t ws_size,
                              hipStream_t stream) {
  const float* q   = (const float*)d_in[0];
  const float* k   = (const float*)d_in[1];
  const float* v   = (const float*)d_in[2];
  const float* Wq  = (const float*)d_in[3];
  const float* Wk  = (const float*)d_in[4];
  const float* Wv  = (const float*)d_in[5];
  const float* Wfc = (const float*)d_in[6];
  const float* lnw = (const float*)d_in[7];
  const float* lnb = (const float*)d_in[8];
  float* out = (float*)d_out;

  const size_t SZ = (size_t)BATCH * OQ * HW;   // floats per intermediate
  float* ws = (float*)d_ws;
  float* qh = ws;
  float* kh = ws + SZ;
  float* vh = ws + 2 * SZ;
  float* ao = ws + 3 * SZ;

  // 1) QKV projections (WMMA f32)
  proj_kernel<<<dim3(HW / 64, OQ / 64, BATCH * 3), 128, 0, stream>>>(
      q, k, v, Wq, Wk, Wv, qh, kh, vh);

  // 2) local windowed attention
  attn_kernel<<<dim3(WDIM / 16, HDIM / 16, BATCH * NHEAD), 256, 0, stream>>>(
      qh, kh, vh, ao);

  // 3) FC + residual + LayerNorm (WMMA f32)
  fc_ln_kernel<<<dim3(HW / 16, BATCH), 256, 0, stream>>>(
      ao, Wfc, q, lnw, lnb, out);
}